// Logic_88905823027892
// MI455X (gfx1250) — compile-verified
//
#include <hip/hip_runtime.h>
#include <hip/hip_bf16.h>
#include <cstdint>

#define IN_DIM   65536
#define OUT_DIM  65536
#define BATCH    1024
#define NG       16

// ---------------------------------------------------------------------------
// CDNA5 async-to-LDS plumbing (builtin confirmed present in rounds 1-3).
// ---------------------------------------------------------------------------
#if defined(__has_builtin)
#  if __has_builtin(__builtin_amdgcn_global_load_async_to_lds_b128)
#    define USE_ASYNC_BUILTIN 1
#  endif
#  if __has_builtin(__builtin_amdgcn_s_wait_asynccnt)
#    define HAVE_WAIT_BUILTIN 1
#  endif
#endif

typedef int   v4i __attribute__((ext_vector_type(4)));
typedef float v2f __attribute__((ext_vector_type(2)));
typedef float v8f __attribute__((ext_vector_type(8)));
typedef __attribute__((address_space(1))) v4i* gv4i_p;   // global
typedef __attribute__((address_space(3))) v4i* lv4i_p;   // LDS

__device__ __forceinline__ void async_16B_to_lds(const float* gsrc, float* lds_dst) {
#if USE_ASYNC_BUILTIN
  __builtin_amdgcn_global_load_async_to_lds_b128(
      (gv4i_p)(uintptr_t)gsrc, (lv4i_p)lds_dst, /*offset=*/0, /*cpol=*/0);
#else
  unsigned lds_addr = (unsigned)(uintptr_t)(__attribute__((address_space(3))) void*)lds_dst;
  asm volatile("global_load_async_to_lds_b128 %0, %1, off"
               :: "v"(lds_addr), "v"(gsrc) : "memory");
#endif
}

__device__ __forceinline__ void wait_async_zero() {
#if HAVE_WAIT_BUILTIN
  __builtin_amdgcn_s_wait_asynccnt(0);
#else
  asm volatile("s_wait_asynccnt 0" ::: "memory");
#endif
}

// ---------------------------------------------------------------------------
// Gate polynomial coefficients: gate_g(a,b) = C0 + CA*a + CB*b + CAB*ab.
// Packed as signed nibbles  [nib0=C0, nib1=CA, nib2=CB, nib3=CAB]  per gate;
// all coefficients are in {-2..2}. Branch-free extraction: shift/mask/
// sign-extend/cvt -- no EXEC writes, no divergence near the WMMAs.
// ---------------------------------------------------------------------------
__device__ __forceinline__ float coefB(int g, int n) {
  // g = 4q + 2*hi + {0,1} varies with the lane half; n in [0,15], columns n>=4
  // read zero nibbles (upper 48 bits are zero).
  constexpr uint64_t P[NG] = {
      0x0000ull, 0x1000ull, 0xF010ull, 0x0010ull,   // 0, ab, a-ab, a
      0xF100ull, 0x0100ull, 0xE110ull, 0xF110ull,   // b-ab, b, a+b-2ab, a+b-ab
      0x1FF1ull, 0x2FF1ull, 0x0F01ull, 0x1F01ull,   // 1-(a+b-ab), 1-(a+b-2ab), 1-b, 1-b+ab
      0x00F1ull, 0x10F1ull, 0xF001ull, 0x0001ull};  // 1-a, 1-a+ab, 1-ab, 1
  const int nib = (int)((P[g] >> (4 * n)) & 0xFull);
  return (float)((nib << 28) >> 28);                // sign-extend nibble -> f32
}

// ---------------------------------------------------------------------------
// Prep via WMMA:  W(65536x4) = softmax(weights)(65536x16) @ C(16x4).
// One wave handles a 16-neuron tile: D(16x16) = sum_q A_q(16x4) @ B_q(4x16)
// using V_WMMA_F32_16X16X4_F32 chained over the K=16 gate axis (4 chunks).
// Columns n>=4 of B are zero, so only D[:,0:4] is meaningful.
// ---------------------------------------------------------------------------
__global__ void __launch_bounds__(256)
lgn_prep_wmma(const float* __restrict__ w, float* __restrict__ Wf) {
  const int lane = threadIdx.x & 31;
  const int wave = threadIdx.x >> 5;
  const int jb   = (blockIdx.x * 8 + wave) * 16;   // neuron tile base
  const int m    = lane & 15;                      // A row / B,D column index
  const int hi   = lane >> 4;                      // K-half select (wave32 split)
  const int row  = jb + m;

  // Softmax of row `row` (lanes L and L+16 both compute it: each carries a
  // different K-slice of the A tile, so both need the full nw[] vector).
  const float4* w4 = (const float4*)(w + row * NG);
  const float4 q0 = w4[0], q1 = w4[1], q2 = w4[2], q3 = w4[3];
  float e[NG] = {q0.x,q0.y,q0.z,q0.w, q1.x,q1.y,q1.z,q1.w,
                 q2.x,q2.y,q2.z,q2.w, q3.x,q3.y,q3.z,q3.w};
  float mx = e[0];
#pragma unroll
  for (int g = 1; g < NG; ++g) mx = fmaxf(mx, e[g]);
  float s = 0.0f;
#pragma unroll
  for (int g = 0; g < NG; ++g) { e[g] = __expf(e[g] - mx); s += e[g]; }
  const float inv = 1.0f / s;
#pragma unroll
  for (int g = 0; g < NG; ++g) e[g] *= inv;

  // Chained K=4 WMMA accumulation over the 16 gates. EXEC is all-ones here
  // (no divergence anywhere before this point; coefB is branch-free).
  v8f acc = {0.f,0.f,0.f,0.f,0.f,0.f,0.f,0.f};
#pragma unroll
  for (int q = 0; q < 4; ++q) {
    v2f A, B;
    // A 16x4 layout: VGPR0 = K(hi?2:0), VGPR1 = K(hi?3:1), row m in lane.
    A.x = hi ? e[4*q + 2] : e[4*q + 0];
    A.y = hi ? e[4*q + 3] : e[4*q + 1];
    // B 4x16 layout: lanes 0-15 carry K rows 0,1; lanes 16-31 carry K rows 2,3.
    B.x = coefB(4*q + 2*hi + 0, m) ;
    B.y = coefB(4*q + 2*hi + 1, m) ;
    acc = __builtin_amdgcn_wmma_f32_16x16x4_f32(
        /*neg_a=*/false, A, /*neg_b=*/false, B,
        /*c_mod=*/(short)0, acc, /*reuse_a=*/false, /*reuse_b=*/false);
  }

  // D layout: VGPR r holds D[M = r + 8*hi][N = m]. Keep columns 0..3.
  if (m < 4) {
#pragma unroll
    for (int r = 0; r < 8; ++r) {
      Wf[(jb + r + 8 * hi) * 4 + m] = acc[r];
    }
  }
}

// ---------------------------------------------------------------------------
// Main: one workgroup per batch row. Stage the whole 256 KB row of x into LDS
// via CDNA5 async copies (possible only because CDNA5 has 320 KB LDS/WGP),
// then each thread produces 64 outputs with LDS gathers + 3 FMAs, writing the
// 256 MB result with non-temporal stores.
// ---------------------------------------------------------------------------
__global__ void __launch_bounds__(1024)
lgn_main(const float* __restrict__ x, const int* __restrict__ a,
         const int* __restrict__ b, const float4* __restrict__ W,
         float* __restrict__ out) {
  extern __shared__ float xrow[];                 // IN_DIM floats = 256 KB
  const int i = blockIdx.x;
  const int t = threadIdx.x;
  const float* __restrict__ xr = x + (size_t)i * IN_DIM;

  // 65536 floats / 1024 threads = 16 x b128 async copies per thread.
#pragma unroll
  for (int k = 0; k < IN_DIM / 4 / 1024; ++k) {
    const int c = (k * 1024 + t) * 4;             // float index, 16B-chunk grain
    async_16B_to_lds(xr + c, xrow + c);
  }
  wait_async_zero();                              // my wave's copies landed
  __syncthreads();                                // everyone's copies landed

  float* __restrict__ orow = out + (size_t)i * OUT_DIM;
#pragma unroll 4
  for (int jj = 0; jj < OUT_DIM / 1024; ++jj) {
    const int j  = jj * 1024 + t;                 // coalesced across the wave
    const int ai = a[j];
    const int bi = b[j];
    const float4 c = W[j];
    const float av = xrow[ai];                    // ds_load_b32 gathers
    const float bv = xrow[bi];
    const float r  = fmaf(c.w, av * bv, fmaf(c.z, bv, fmaf(c.y, av, c.x)));
    __builtin_nontemporal_store(r, orow + j);     // stream out, keep L2 for coefs
  }
}

// ---------------------------------------------------------------------------
extern "C" void kernel_launch(void* const* d_in, const int* in_sizes, int n_in,
                              void* d_out, int out_size, void* d_ws, size_t ws_size,
                              hipStream_t stream) {
  const float* x = (const float*)d_in[0];   // (BATCH, IN_DIM) f32
  const float* w = (const float*)d_in[1];   // (OUT_DIM, 16) f32
  const int*   a = (const int*)d_in[2];     // (OUT_DIM,) int (harness: int32)
  const int*   b = (const int*)d_in[3];     // (OUT_DIM,) int
  float* Wf = (float*)d_ws;                 // 65536 * 4 f32 = 1 MB scratch

  // W = softmax(weights) @ C, on the WMMA pipe: 4096 16-neuron tiles.
  lgn_prep_wmma<<<OUT_DIM / 16 / 8, 256, 0, stream>>>(w, Wf);
  lgn_main<<<BATCH, 1024, IN_DIM * sizeof(float), stream>>>(x, a, b,
                                                            (const float4*)Wf,
                                                            (float*)d_out);
}